// SelMul_71210557768253
// MI455X (gfx1250) — compile-verified
//
#include <hip/hip_runtime.h>

// Problem constants (match reference): B=256, D=512, OUT=1024, H=(D+1)*D/2
#define B_SZ   256
#define D_SZ   512
#define OUT_SZ 1024
#define H_SZ   ((D_SZ + 1) * D_SZ / 2)   // 131328

typedef __attribute__((ext_vector_type(2))) float v2f;
typedef __attribute__((ext_vector_type(8))) float v8f;

// C[256,1024] = P[256,H] * W^T,  P[b,h(i,j)] = x[b,i]*x[b,j]  (i<=j, row-major triu)
// Block: 128 threads = 4 waves; block tile = 16(M) x 128(N).
// Each wave computes TWO 16x16 N-tiles (columns o_base+wave*32 .. +31) that share
// one A fragment per K-group, accumulated with V_WMMA_F32_16X16X4_F32.
__global__ __launch_bounds__(128)
void trivec_gemm_f32wmma(const float* __restrict__ x,
                         const float* __restrict__ W,
                         float* __restrict__ out) {
    __shared__ float xs[16 * 513];   // x tile, stride 513 kills LDS bank conflicts

    const int m_base = blockIdx.x * 16;
    const int o_base = blockIdx.y * 128;
    const int tid    = threadIdx.x;

    // Stage x[m_base..m_base+15, :] into LDS (coalesced)
    for (int idx = tid; idx < 16 * D_SZ; idx += 128) {
        const int m = idx >> 9;
        const int c = idx & (D_SZ - 1);
        xs[m * 513 + c] = x[(size_t)(m_base + m) * D_SZ + c];
    }
    __syncthreads();

    const int  wave = tid >> 5;
    const int  lane = tid & 31;
    const int  l16  = lane & 15;
    const bool hi   = lane >= 16;
    const int  kofs = hi ? 2 : 0;    // K slots {0,1} for lanes 0-15, {2,3} for 16-31

    const float* __restrict__ xrow = xs + l16 * 513;   // A row m = l16

    // Two uniform W bases (SGPR pairs) + one shared per-lane u32 element offset.
    const float* __restrict__ wb0 = W + (size_t)(o_base + wave * 32) * H_SZ;
    const float* __restrict__ wb1 = wb0 + (size_t)16 * H_SZ;
    const unsigned lane_off = (unsigned)l16 * (unsigned)H_SZ + (unsigned)kofs;

    v8f acc0 = {0.f,0.f,0.f,0.f,0.f,0.f,0.f,0.f};
    v8f acc1 = {0.f,0.f,0.f,0.f,0.f,0.f,0.f,0.f};

    int i = 0, j = 0, h = 0;          // uniform triangle walkers
    while (h < H_SZ) {
        const int ng = (D_SZ - j) >> 2;      // clean K-groups left in row i
        if (ng > 0) {
            const float xi = xrow[i];        // one LDS dword per row
            const float* xp = xrow + j + kofs;
            __builtin_prefetch(wb0 + lane_off + h + 2048, 0, 1);
            __builtin_prefetch(wb1 + lane_off + h + 2048, 0, 1);
            unsigned voff = lane_off + (unsigned)h;
            #pragma unroll 2
            for (int g = 0; g < ng; ++g) {
                const float2 xv = *reinterpret_cast<const float2*>(xp);
                v2f a;
                a.x = xi * xv.x;
                a.y = xi * xv.y;
                const float2 w0 = *reinterpret_cast<const float2*>(wb0 + voff);
                const float2 w1 = *reinterpret_cast<const float2*>(wb1 + voff);
                v2f b0; b0.x = w0.x; b0.y = w0.y;
                v2f b1; b1.x = w1.x; b1.y = w1.y;
                acc0 = __builtin_amdgcn_wmma_f32_16x16x4_f32(
                    false, a, false, b0, (short)0, acc0, false, false);
                acc1 = __builtin_amdgcn_wmma_f32_16x16x4_f32(
                    false, a, false, b1, (short)0, acc1, false, false);
                xp   += 4;
                voff += 4;
            }
            j += ng << 2;
            h += ng << 2;
            if (h >= H_SZ) break;
        }
        if (j == D_SZ) { ++i; j = i; continue; }

        // Row-boundary K-group (<1.7% of groups): generic 4-step with wrap.
        {
            int ii[4], jj[4];
            #pragma unroll
            for (int k = 0; k < 4; ++k) {
                ii[k] = i; jj[k] = j;
                if (++j == D_SZ) { ++i; j = i; }
            }
            const int ia = hi ? ii[2] : ii[0];
            const int ja = hi ? jj[2] : jj[0];
            const int ib = hi ? ii[3] : ii[1];
            const int jb = hi ? jj[3] : jj[1];
            v2f a;
            a.x = xrow[ia] * xrow[ja];
            a.y = xrow[ib] * xrow[jb];
            const unsigned voff = lane_off + (unsigned)h;
            const float2 w0 = *reinterpret_cast<const float2*>(wb0 + voff);
            const float2 w1 = *reinterpret_cast<const float2*>(wb1 + voff);
            v2f b0; b0.x = w0.x; b0.y = w0.y;
            v2f b1; b1.x = w1.x; b1.y = w1.y;
            acc0 = __builtin_amdgcn_wmma_f32_16x16x4_f32(
                false, a, false, b0, (short)0, acc0, false, false);
            acc1 = __builtin_amdgcn_wmma_f32_16x16x4_f32(
                false, a, false, b1, (short)0, acc1, false, false);
            h += 4;
        }
    }

    // C/D layout: VGPR v -> M = v + (hi?8:0), N = lane&15 within each 16x16 tile.
    const int row0 = hi ? 8 : 0;
    float* out0 = out + (size_t)(m_base + row0) * OUT_SZ + o_base + wave * 32 + l16;
    float* out1 = out0 + 16;
    #pragma unroll
    for (int v = 0; v < 8; ++v) {
        out0[(size_t)v * OUT_SZ] = acc0[v];
        out1[(size_t)v * OUT_SZ] = acc1[v];
    }
}

extern "C" void kernel_launch(void* const* d_in, const int* in_sizes, int n_in,
                              void* d_out, int out_size, void* d_ws, size_t ws_size,
                              hipStream_t stream) {
    const float* x = (const float*)d_in[0];   // (256, 512)  f32
    const float* W = (const float*)d_in[1];   // (1024, 131328) f32
    float* out = (float*)d_out;               // (256, 1024) f32

    dim3 grid(B_SZ / 16, OUT_SZ / 128);       // (16, 8) -> 128 blocks, 512 waves
    trivec_gemm_f32wmma<<<grid, 128, 0, stream>>>(x, W, out);
}